// linears_9715216023980
// MI455X (gfx1250) — compile-verified
//
#include <hip/hip_runtime.h>

typedef float v2f __attribute__((ext_vector_type(2)));
typedef float v8f __attribute__((ext_vector_type(8)));
typedef unsigned int v4u __attribute__((ext_vector_type(4)));
typedef int v8i __attribute__((ext_vector_type(8)));
typedef int v4i __attribute__((ext_vector_type(4)));

#define T_DIM 512
#define IN_F  1024
#define OUT_F 1024
#define BM    128           // M (sequence) tile per block
#define BN    128           // N (out-feature) tile per block
#define KT    16            // K stage depth (TDM tile_dim0)
#define ROWW  20            // KT + 4 pad dwords (TDM pad: interval 16dw, amount 4dw)
#define NSTAGES (IN_F / KT) // 64

// ---- Tensor Data Mover: DMA one [tile_rows x KT] f32 tile (row stride IN_F)
// from global memory into LDS at lds_addr, padding each 16-dword row with
// 4 dwords so the LDS row stride is 20 dwords (bank-conflict-free reads).
__device__ __forceinline__ void tdm_load_tile(unsigned lds_addr, const float* gaddr)
{
    unsigned long long ga = (unsigned long long)(uintptr_t)gaddr;

    // D# group 0 (4 dwords): count=1 | lds_addr | global_addr[56:0] | type=2
    v4u g0;
    g0.x = 1u;                                             // count=1, user desc
    g0.y = lds_addr;                                       // [63:32] lds_addr
    g0.z = (unsigned)(ga & 0xffffffffull);                 // [95:64] gaddr lo
    g0.w = (unsigned)((ga >> 32) & 0x1ffffffull)           // [120:96] gaddr hi
         | (2u << 30);                                     // [127:126] type=2

    // D# group 1 (8 dwords)
    const unsigned td0 = IN_F;       // tensor_dim0 (elements) - never OOB here
    const unsigned td1 = 1u << 20;   // tensor_dim1 large - never OOB
    v8i g1;
    g1[0] = (int)((2u << 16)         // data_size = 2 -> 4 bytes
                | (1u << 20)         // pad_enable
                | (3u << 22)         // pad_interval code 3 -> every 16 dwords
                | (3u << 25));       // pad_amount  code 3 -> 4 dwords
    g1[1] = (int)((td0 & 0xffffu) << 16);                  // atomic_addr=0 | td0 lo16
    g1[2] = (int)((td0 >> 16) | ((td1 & 0xffffu) << 16));  // td0 hi16 | td1 lo16
    g1[3] = (int)((td1 >> 16) | ((unsigned)KT << 16));     // td1 hi16 | tile_dim0
    g1[4] = (int)BM;                                       // tile_dim1=128 | tile_dim2=0
    g1[5] = (int)IN_F;                                     // tensor_dim0_stride lo32
    g1[6] = 0;                                             // stride0 hi | stride1 lo
    g1[7] = 0;                                             // stride1 hi

    v4i z4 = {0, 0, 0, 0};                                 // groups 2/3: 2-D tile
#if __clang_major__ >= 23
    v8i z8 = {0, 0, 0, 0, 0, 0, 0, 0};
    __builtin_amdgcn_tensor_load_to_lds(g0, g1, z4, z4, z8, 0);
#else
    __builtin_amdgcn_tensor_load_to_lds(g0, g1, z4, z4, 0);
#endif
}

__global__ __launch_bounds__(256) void grouped_linear_wmma_f32(
    const float* __restrict__ x,     // [B, T, IN]
    const float* __restrict__ W,     // [B, OUT, IN]
    const float* __restrict__ bias,  // [B, OUT]
    float* __restrict__ out)         // [B, T, OUT]
{
    __shared__ float xs[2][BM][ROWW];   // double-buffered x tile
    __shared__ float ws[2][BN][ROWW];   // double-buffered W tile

    const int g  = blockIdx.z;
    const int n0 = blockIdx.x * BN;
    const int m0 = blockIdx.y * BM;

    const int tid   = threadIdx.x;      // 0..255 (8 waves, wave32)
    const int wave  = tid >> 5;
    const int lane  = tid & 31;
    const int lrow  = lane & 15;        // fragment row/col within 16
    const int lhalf = lane >> 4;        // K-pair selector

    const int wm = (wave & 1) * 64;     // wave M offset in block tile
    const int wn = (wave >> 1) * 32;    // wave N offset in block tile

    const float* xg = x + (size_t)g * T_DIM * IN_F + (size_t)m0 * IN_F;
    const float* Wg = W + (size_t)g * OUT_F * IN_F + (size_t)n0 * IN_F;

    const unsigned xs_lds[2] = { (unsigned)(uintptr_t)&xs[0][0][0],
                                 (unsigned)(uintptr_t)&xs[1][0][0] };
    const unsigned ws_lds[2] = { (unsigned)(uintptr_t)&ws[0][0][0],
                                 (unsigned)(uintptr_t)&ws[1][0][0] };

    v8f acc[4][2];
    #pragma unroll
    for (int tm = 0; tm < 4; ++tm)
        #pragma unroll
        for (int tn = 0; tn < 2; ++tn)
            acc[tm][tn] = (v8f)0.0f;

    // Prologue: wave 0 kicks off the DMA for stage 0.
    if (wave == 0) {
        tdm_load_tile(xs_lds[0], xg);
        tdm_load_tile(ws_lds[0], Wg);
    }

    int buf = 0;
    for (int s = 0; s < NSTAGES; ++s) {
        if (wave == 0)
            __builtin_amdgcn_s_wait_tensorcnt(0);   // stage-s tiles landed in LDS
        __syncthreads();                            // publish to all waves

        // Overlap: issue stage s+1 DMA into the other buffer while computing.
        if (wave == 0 && (s + 1) < NSTAGES) {
            const int k1 = (s + 1) * KT;
            tdm_load_tile(xs_lds[buf ^ 1], xg + k1);
            tdm_load_tile(ws_lds[buf ^ 1], Wg + k1);
        }

        const float (*xsb)[ROWW] = xs[buf];
        const float (*wsb)[ROWW] = ws[buf];

        #pragma unroll
        for (int kk = 0; kk < KT; kk += 4) {
            const int kb = kk + lhalf * 2;          // lane-half selects K pair

            v2f afrag[4];
            #pragma unroll
            for (int tm = 0; tm < 4; ++tm) {
                const float* p = &xsb[wm + tm * 16 + lrow][kb];
                afrag[tm].x = p[0];
                afrag[tm].y = p[1];
            }
            v2f bfrag[2];
            #pragma unroll
            for (int tn = 0; tn < 2; ++tn) {
                const float* p = &wsb[wn + tn * 16 + lrow][kb];
                bfrag[tn].x = p[0];
                bfrag[tn].y = p[1];
            }

            #pragma unroll
            for (int tm = 0; tm < 4; ++tm)
                #pragma unroll
                for (int tn = 0; tn < 2; ++tn)
                    acc[tm][tn] = __builtin_amdgcn_wmma_f32_16x16x4_f32(
                        false, afrag[tm], false, bfrag[tn],
                        (short)0, acc[tm][tn], false, false);
        }
        buf ^= 1;
    }

    // ---- epilogue: add bias, store fp32 ----
    float bv[2];
    #pragma unroll
    for (int tn = 0; tn < 2; ++tn)
        bv[tn] = bias[(size_t)g * OUT_F + n0 + wn + tn * 16 + lrow];

    float* og = out + (size_t)g * T_DIM * OUT_F;
    #pragma unroll
    for (int tm = 0; tm < 4; ++tm) {
        #pragma unroll
        for (int tn = 0; tn < 2; ++tn) {
            const int n = n0 + wn + tn * 16 + lrow;
            #pragma unroll
            for (int v = 0; v < 8; ++v) {
                const int m = m0 + wm + tm * 16 + lhalf * 8 + v;
                og[(size_t)m * OUT_F + n] = acc[tm][tn][v] + bv[tn];
            }
        }
    }
}

extern "C" void kernel_launch(void* const* d_in, const int* in_sizes, int n_in,
                              void* d_out, int out_size, void* d_ws, size_t ws_size,
                              hipStream_t stream) {
    (void)in_sizes; (void)n_in; (void)out_size; (void)d_ws; (void)ws_size;
    const float* x  = (const float*)d_in[0];   // [64, 512, 1024]
    const float* W  = (const float*)d_in[1];   // [64, 1024, 1024]
    const float* b  = (const float*)d_in[2];   // [64, 1024]
    float* out      = (float*)d_out;           // [64, 512, 1024]

    dim3 grid(OUT_F / BN, T_DIM / BM, 64);     // (8, 4, 64)
    dim3 block(256);
    grouped_linear_wmma_f32<<<grid, block, 0, stream>>>(x, W, b, out);
}